// ResD1LayerTopK_6176162972423
// MI455X (gfx1250) — compile-verified
//
#include <hip/hip_runtime.h>
#include <math.h>

#define Bn 1024
#define Dn 256
#define Hn 256
#define Kn 8
#define LN_EPS 1e-5f

typedef float v2f __attribute__((ext_vector_type(2)));
typedef float v8f __attribute__((ext_vector_type(8)));

// ---------------------------------------------------------------------------
// One wave computes a 16x16 f32 tile of  A * Bm^T  using V_WMMA_F32_16X16X4_F32.
// A is row-major [.., lda], Bm is row-major [.., ldb] (we multiply by its
// transpose, so "columns" of the B matrix are rows of Bm -> contiguous loads).
// Per ISA 32-bit A 16x4 layout: lane (half*16+l16) holds A[m=l16][k0+2*half+r],
// r=0,1 -> a contiguous float2. B 4x16 mirrors it. C/D: VGPR r holds row
// (r + 8*half), col l16.
// ---------------------------------------------------------------------------
__device__ __forceinline__ v8f wmma_tile_ABt(const float* __restrict__ A,
                                             const float* __restrict__ Bm,
                                             int lda, int ldb,
                                             int rowA, int rowB,
                                             int kdim, int lane) {
  const int l16 = lane & 15;
  const int half = lane >> 4;
  const float* ap = A + (size_t)(rowA + l16) * lda + 2 * half;
  const float* bp = Bm + (size_t)(rowB + l16) * ldb + 2 * half;
  v8f c = {0.f, 0.f, 0.f, 0.f, 0.f, 0.f, 0.f, 0.f};
  for (int k = 0; k < kdim; k += 4) {
    v2f a = *reinterpret_cast<const v2f*>(ap + k);
    v2f b = *reinterpret_cast<const v2f*>(bp + k);
    c = __builtin_amdgcn_wmma_f32_16x16x4_f32(false, a, false, b,
                                              (short)0, c, false, false);
  }
  return c;
}

// ---- K1: per-row squared norm + temperature tau -----------------------------
__global__ void k_norm_tau(const float* __restrict__ x,
                           const float* __restrict__ Wtau,
                           const float* __restrict__ btau,
                           float* __restrict__ n2, float* __restrict__ tau) {
  __shared__ float s0[256];
  __shared__ float s1[256];
  const int i = blockIdx.x, t = threadIdx.x;
  const float xv = x[i * Dn + t];
  s0[t] = xv * xv;
  s1[t] = xv * Wtau[t];
  __syncthreads();
  for (int off = 128; off > 0; off >>= 1) {
    if (t < off) { s0[t] += s0[t + off]; s1[t] += s1[t + off]; }
    __syncthreads();
  }
  if (t == 0) {
    n2[i] = s0[0];
    const float z = s1[0] + btau[0];
    const float sp = (z > 20.f) ? z : log1pf(expf(z));  // softplus
    tau[i] = fmaxf(sp, 0.01f) + 1.0f;
  }
}

// ---- K2: dist^2 = |xi|^2 + |xj|^2 - 2 * (X X^T)  via f32 WMMA ---------------
__global__ void k_dist2(const float* __restrict__ x,
                        const float* __restrict__ n2,
                        float* __restrict__ dist2) {
  const int wave = (blockIdx.x * blockDim.x + threadIdx.x) >> 5;
  const int lane = threadIdx.x & 31;
  const int ti = wave >> 6;          // 64 row tiles
  const int tj = wave & 63;          // 64 col tiles
  const int rowA = ti * 16, rowB = tj * 16;
  v8f c = wmma_tile_ABt(x, x, Dn, Dn, rowA, rowB, Dn, lane);
  const int l16 = lane & 15, half = lane >> 4;
  const int n = rowB + l16;
  const float n2n = n2[n];
#pragma unroll
  for (int r = 0; r < 8; ++r) {
    const int m = rowA + r + half * 8;
    const float d = n2[m] + n2n - 2.0f * c[r];
    dist2[(size_t)m * Bn + n] = fmaxf(d, 0.0f);   // clamp: keeps u64 key order valid
  }
}

// ---- K3: per-row top-8 smallest (ties -> smaller index, matching lax.top_k) -
__global__ void k_topk(const float* __restrict__ dist2, int* __restrict__ topk) {
  __shared__ float row[Bn];
  __shared__ unsigned long long red[256];
  const int i = blockIdx.x, t = threadIdx.x;
  for (int j = t; j < Bn; j += 256) row[j] = dist2[(size_t)i * Bn + j];
  __syncthreads();
  for (int k = 0; k < Kn; ++k) {
    unsigned long long best = ~0ull;
    for (int j = t; j < Bn; j += 256) {
      const unsigned long long key =
          ((unsigned long long)__float_as_uint(row[j]) << 32) | (unsigned)j;
      best = (key < best) ? key : best;
    }
    red[t] = best;
    __syncthreads();
    for (int off = 128; off > 0; off >>= 1) {
      if (t < off) {
        const unsigned long long o = red[t + off];
        if (o < red[t]) red[t] = o;
      }
      __syncthreads();
    }
    const int jmin = (int)(red[0] & 0xffffffffu);
    if (t == 0) {
      topk[i * Kn + k] = jmin;
      row[jmin] = __int_as_float(0x7f800000);  // +inf, remove from next pass
    }
    __syncthreads();
  }
}

// ---- K4: dbar[i][d] = mean_k |x[i][d] - x[nbr[k]][d]| -----------------------
__global__ void k_dbar(const float* __restrict__ x, const int* __restrict__ topk,
                       float* __restrict__ dbar) {
  __shared__ int nb[Kn];
  const int i = blockIdx.x, t = threadIdx.x;
  if (t < Kn) nb[t] = topk[i * Kn + t];
  __syncthreads();
  const float xi = x[i * Dn + t];
  float acc = 0.f;
#pragma unroll
  for (int k = 0; k < Kn; ++k) acc += fabsf(xi - x[(size_t)nb[k] * Dn + t]);
  dbar[i * Dn + t] = acc * (1.0f / Kn);
}

// ---- K5: hk = (dbar @ W_diff^T + b_diff) / tau  via f32 WMMA ----------------
__global__ void k_hk(const float* __restrict__ dbar,
                     const float* __restrict__ Wdiff,
                     const float* __restrict__ bdiff,
                     const float* __restrict__ tau, float* __restrict__ hk) {
  const int wave = (blockIdx.x * blockDim.x + threadIdx.x) >> 5;
  const int lane = threadIdx.x & 31;
  const int ti = wave >> 4;          // 64 row tiles
  const int tj = wave & 15;          // 16 col tiles
  v8f c = wmma_tile_ABt(dbar, Wdiff, Dn, Dn, ti * 16, tj * 16, Dn, lane);
  const int l16 = lane & 15, half = lane >> 4;
  const int n = tj * 16 + l16;
  const float bb = bdiff[n];
#pragma unroll
  for (int r = 0; r < 8; ++r) {
    const int m = ti * 16 + r + half * 8;
    hk[(size_t)m * Hn + n] = (c[r] + bb) / tau[m];
  }
}

// ---- K6: y = relu(hk @ W_agg^T + b_agg) + (x @ W_res^T + b_res) -------------
__global__ void k_fused_out(const float* __restrict__ hk,
                            const float* __restrict__ Wagg,
                            const float* __restrict__ bagg,
                            const float* __restrict__ x,
                            const float* __restrict__ Wres,
                            const float* __restrict__ bres,
                            float* __restrict__ y) {
  const int wave = (blockIdx.x * blockDim.x + threadIdx.x) >> 5;
  const int lane = threadIdx.x & 31;
  const int ti = wave >> 4;
  const int tj = wave & 15;
  v8f ch = wmma_tile_ABt(hk, Wagg, Hn, Hn, ti * 16, tj * 16, Hn, lane);
  v8f cx = wmma_tile_ABt(x,  Wres, Dn, Dn, ti * 16, tj * 16, Dn, lane);
  const int l16 = lane & 15, half = lane >> 4;
  const int n = tj * 16 + l16;
  const float ba = bagg[n], br = bres[n];
#pragma unroll
  for (int r = 0; r < 8; ++r) {
    const int m = ti * 16 + r + half * 8;
    y[(size_t)m * Hn + n] = fmaxf(ch[r] + ba, 0.0f) + cx[r] + br;
  }
}

// ---- K7: LayerNorm over H with gamma/beta -----------------------------------
__global__ void k_layernorm(const float* __restrict__ y,
                            const float* __restrict__ gamma,
                            const float* __restrict__ beta,
                            float* __restrict__ out) {
  __shared__ float s[256];
  __shared__ float mu_s, rs_s;
  const int i = blockIdx.x, t = threadIdx.x;
  const float v = y[(size_t)i * Hn + t];
  s[t] = v;
  __syncthreads();
  for (int off = 128; off > 0; off >>= 1) {
    if (t < off) s[t] += s[t + off];
    __syncthreads();
  }
  if (t == 0) mu_s = s[0] * (1.0f / Hn);
  __syncthreads();
  const float d = v - mu_s;
  s[t] = d * d;
  __syncthreads();
  for (int off = 128; off > 0; off >>= 1) {
    if (t < off) s[t] += s[t + off];
    __syncthreads();
  }
  if (t == 0) rs_s = rsqrtf(s[0] * (1.0f / Hn) + LN_EPS);
  __syncthreads();
  out[(size_t)i * Hn + t] = d * rs_s * gamma[t] + beta[t];
}

// ---------------------------------------------------------------------------
extern "C" void kernel_launch(void* const* d_in, const int* in_sizes, int n_in,
                              void* d_out, int out_size, void* d_ws, size_t ws_size,
                              hipStream_t stream) {
  const float* x     = (const float*)d_in[0];
  const float* Wdiff = (const float*)d_in[1];
  const float* bdiff = (const float*)d_in[2];
  const float* Wtau  = (const float*)d_in[3];
  const float* btau  = (const float*)d_in[4];
  const float* Wagg  = (const float*)d_in[5];
  const float* bagg  = (const float*)d_in[6];
  const float* Wres  = (const float*)d_in[7];
  const float* bres  = (const float*)d_in[8];
  const float* gamma = (const float*)d_in[9];
  const float* beta  = (const float*)d_in[10];
  float* out = (float*)d_out;

  char* ws = (char*)d_ws;
  size_t off = 0;
  float* dist2 = (float*)(ws + off); off += (size_t)Bn * Bn * sizeof(float);
  float* n2    = (float*)(ws + off); off += (size_t)Bn * sizeof(float);
  float* tau   = (float*)(ws + off); off += (size_t)Bn * sizeof(float);
  int*   topk  = (int*)  (ws + off); off += (size_t)Bn * Kn * sizeof(int);
  float* dbar  = (float*)(ws + off); off += (size_t)Bn * Dn * sizeof(float);
  float* hk    = (float*)(ws + off); off += (size_t)Bn * Hn * sizeof(float);
  float* ybuf  = (float*)(ws + off); off += (size_t)Bn * Hn * sizeof(float);
  (void)ws_size; (void)in_sizes; (void)n_in; (void)out_size;

  k_norm_tau<<<Bn, 256, 0, stream>>>(x, Wtau, btau, n2, tau);
  // 64x64 = 4096 16x16 tiles, 4 waves (128 threads) per block
  k_dist2<<<1024, 128, 0, stream>>>(x, n2, dist2);
  k_topk<<<Bn, 256, 0, stream>>>(dist2, topk);
  k_dbar<<<Bn, 256, 0, stream>>>(x, topk, dbar);
  // 64x16 = 1024 tiles
  k_hk<<<256, 128, 0, stream>>>(dbar, Wdiff, bdiff, tau, hk);
  k_fused_out<<<256, 128, 0, stream>>>(hk, Wagg, bagg, x, Wres, bres, ybuf);
  k_layernorm<<<Bn, 256, 0, stream>>>(ybuf, gamma, beta, out);
}